// Translator_87746181857384
// MI455X (gfx1250) — compile-verified
//
#include <hip/hip_runtime.h>
#include <hip/hip_bf16.h>

// Problem constants from the reference.
#define B_ 16
#define S_ 4096
#define D_ 512
#define U_ 128

#define GLOBAL_AS __attribute__((address_space(1)))
#define LDS_AS    __attribute__((address_space(3)))

typedef int v4i __attribute__((ext_vector_type(4)));

#if __has_builtin(__builtin_amdgcn_global_load_async_to_lds_b128) && \
    __has_builtin(__builtin_amdgcn_global_store_async_from_lds_b128)
#define HAVE_ASYNC_LDS 1
#else
#define HAVE_ASYNC_LDS 0
#endif

static __device__ __forceinline__ void wait_asynccnt0() {
#if __has_builtin(__builtin_amdgcn_s_wait_asynccnt)
  __builtin_amdgcn_s_wait_asynccnt(0);
#else
  asm volatile("s_wait_asynccnt 0" ::: "memory");
#endif
}

// One block per (b, u) segment row. 128 threads = 4 wave32.
// Valid rows are copied global -> LDS -> global via the CDNA5 async path
// (ASYNCcnt-tracked, no VGPR staging); invalid rows are zero-filled.
__global__ __launch_bounds__(128) void Translator_gather_first_tokens(
    const float* __restrict__ x,        // [B, S, D]
    const int*   __restrict__ sections, // [B, U]
    float*       __restrict__ res1,     // [B, U, D]
    float*       __restrict__ mask) {   // [B, 1, U] as 0.0/1.0
  __shared__ int   sec[U_];
  __shared__ float row[D_];

  const int u = blockIdx.x;
  const int b = blockIdx.y;
  const int t = threadIdx.x;

  // Stage this batch's section lengths once; L2-resident after first block.
  sec[t] = sections[b * U_ + t];
  __syncthreads();

  // Exclusive prefix sum for our segment (uniform, broadcast LDS reads).
  int off = 0;
  for (int j = 0; j < u; ++j) off += sec[j];
  const bool valid = sec[u] > 0;

  float* dst = res1 + ((size_t)b * U_ + u) * D_;

  if (valid) {
    const float* src = x + ((size_t)b * S_ + off) * D_;
#if HAVE_ASYNC_LDS
    // 128 lanes x 16B = full 2048B row in one async b128 per lane.
    __builtin_amdgcn_global_load_async_to_lds_b128(
        (GLOBAL_AS v4i*)(src + t * 4),
        (LDS_AS   v4i*)(&row[t * 4]), 0, 0);
    // Each lane stores from the LDS bytes its own load wrote -> per-wave
    // ASYNCcnt wait is sufficient ordering; no workgroup barrier needed.
    wait_asynccnt0();
    __builtin_amdgcn_global_store_async_from_lds_b128(
        (GLOBAL_AS v4i*)(dst + t * 4),
        (LDS_AS   v4i*)(&row[t * 4]), 0, 0);
    if (t == 0) mask[b * U_ + u] = (row[0] != 0.0f) ? 1.0f : 0.0f;
    wait_asynccnt0();
#else
    const float4 v = *(const float4*)(src + t * 4);
    *(float4*)(dst + t * 4) = v;
    if (t == 0) mask[b * U_ + u] = (v.x != 0.0f) ? 1.0f : 0.0f;
#endif
  } else {
    *(float4*)(dst + t * 4) = make_float4(0.f, 0.f, 0.f, 0.f);
    if (t == 0) mask[b * U_ + u] = 0.0f;
  }
}

extern "C" void kernel_launch(void* const* d_in, const int* in_sizes, int n_in,
                              void* d_out, int out_size, void* d_ws, size_t ws_size,
                              hipStream_t stream) {
  (void)in_sizes; (void)n_in; (void)d_ws; (void)ws_size; (void)out_size;

  const float* x        = (const float*)d_in[0]; // [16, 4096, 512] f32
  const int*   sections = (const int*)d_in[1];   // [16, 128] i32

  float* res1 = (float*)d_out;                          // [16, 128, 512]
  float* mask = (float*)d_out + (size_t)B_ * U_ * D_;   // [16, 1, 128]

  dim3 grid(U_, B_);
  Translator_gather_first_tokens<<<grid, 128, 0, stream>>>(x, sections, res1, mask);
}